// Embeded_gate_16673063043391
// MI455X (gfx1250) — compile-verified
//
#include <hip/hip_runtime.h>
#include <stdint.h>

// Native vector types (HIP's float4 is a struct; builtins want ext_vector).
typedef float    f32x4 __attribute__((ext_vector_type(4)));
typedef uint32_t u32x4 __attribute__((ext_vector_type(4)));
typedef int32_t  i32x4 __attribute__((ext_vector_type(4)));
typedef int32_t  i32x8 __attribute__((ext_vector_type(8)));

#define W        512           // plane width == height
#define NPLANES  (4 * 32)      // N*C
#define ROWS_OUT 2             // output rows per workgroup
#define NROWS_IN 4             // staged input rows per workgroup (h0-1 .. h0+2)

__global__ __launch_bounds__(256) void embeded_gate_tdm_kernel(
    const float* __restrict__ x,     // (4,32,512,512)
    const float* __restrict__ bias,  // (32,)
    float* __restrict__ out)         // G1 | G2 | G3 concatenated
{
    __shared__ float sm[NROWS_IN][W];   // 8 KB: rows h0-1 .. h0+2 (edge-clamped)

    const int tid   = threadIdx.x;
    const int r0    = blockIdx.x * ROWS_OUT;   // first output row, 0..65535
    const int h0    = r0 & (W - 1);            // row within plane (even)
    const int plane = r0 >> 9;                 // n*32 + c

    // ---- Stage 4 input rows into LDS via the Tensor Data Mover ----
    // Waves 0..3 each DMA one 512x1 fp32 row (2 KB). EXEC is ignored by
    // TENSOR_LOAD_TO_LDS, but the wave-uniform branch means only 4 waves
    // reach the instruction.
    const int wid = tid >> 5;
    if (wid < NROWS_IN) {
        int hh = h0 - 1 + wid;
        hh = hh < 0 ? 0 : (hh > W - 1 ? W - 1 : hh);   // clamp at plane edges
        const uint64_t ga = (uint64_t)(uintptr_t)x +
                            ((uint64_t)(plane * W + hh)) * (uint64_t)(W * 4);
        const uint32_t lds = (uint32_t)(uintptr_t)(&sm[wid][0]);

        // D# group 0 (128b): count=1 | lds_addr | global_addr[56:0] | type=2
        u32x4 g0;
        g0[0] = 1u;                                     // count=1, user D#
        g0[1] = lds;                                    // lds_addr (bytes)
        g0[2] = (uint32_t)ga;                           // global_addr[31:0]
        g0[3] = (uint32_t)((ga >> 32) & 0x01FFFFFFu)    // global_addr[56:32]
              | 0x80000000u;                            // type=2 ("image")

        // D# group 1 (256b): 2D tensor/tile = 512x1 fp32, stride 512.
        i32x8 g1;
        g1[0] = 0x00020000;          // data_size=2 (4B); no mask/pad/iterate
        g1[1] = (int)(W << 16);      // tensor_dim0 = 512 (low16 in [31:16])
        g1[2] = (int)(1 << 16);      // tensor_dim1 = 1
        g1[3] = (int)(W << 16);      // tile_dim0   = 512
        g1[4] = 1;                   // tile_dim1 = 1, tile_dim2 = 0
        g1[5] = W;                   // tensor_dim0_stride = 512
        g1[6] = 0;                   // stride hi / dim1_stride (unused)
        g1[7] = 0;

        i32x4 z4 = {0, 0, 0, 0};     // groups 2/3 unused (2D tensor)
#if defined(__clang_major__) && (__clang_major__ >= 23)
        i32x8 z8 = {0, 0, 0, 0, 0, 0, 0, 0};
        __builtin_amdgcn_tensor_load_to_lds(g0, g1, z4, z4, z8, 0);
#else
        __builtin_amdgcn_tensor_load_to_lds(g0, g1, z4, z4, 0);
#endif
    }
    __builtin_amdgcn_s_wait_tensorcnt(0);   // wave's own TDM ops complete
    __syncthreads();                        // LDS visible to all 8 waves

    // ---- Compute: each thread -> 4 pixels x 3 outputs ----
    const int lr = tid >> 7;        // which of the 2 output rows (0/1)
    const int t  = tid & 127;       // 128 threads per row, 4 px each
    const int w0 = t * 4;
    const int h  = h0 + lr;

    const float* __restrict__ prevR = sm[lr];       // x[h-1] (clamped)
    const float* __restrict__ curR  = sm[lr + 1];   // x[h]
    const float* __restrict__ nextR = sm[lr + 2];   // x[h+1] (clamped)

    const float bv = bias[plane & 31];

    const int wl = (w0 == 0) ? 0 : (w0 - 1);        // safe left-halo index
    const float cm1 = curR[wl];
    const float pm1 = prevR[wl];
    const float nm1 = nextR[wl];
    const f32x4 cv = *(const f32x4*)&curR[w0];      // x[h][w0..w0+3]
    const f32x4 pv = *(const f32x4*)&prevR[w0];
    const f32x4 nv = *(const f32x4*)&nextR[w0];

    const bool hp   = (h > 0);       // G1 valid rows
    const bool hn   = (h < W - 1);   // G3 valid rows
    const bool wok0 = (w0 > 0);      // w>=1 for the first pixel in the quad

#define GATE(u, v) (__expf(-((u) - (v)) * ((u) - (v))) + bv)

    f32x4 o1, o2, o3;
    // G1[h][w] = exp(-(x[h-1][w-1]-x[h][w])^2)+b  (h>=1, w>=1)
    o1.x = (hp && wok0) ? GATE(pm1,  cv.x) : bv;
    o1.y =  hp          ? GATE(pv.x, cv.y) : bv;
    o1.z =  hp          ? GATE(pv.y, cv.z) : bv;
    o1.w =  hp          ? GATE(pv.z, cv.w) : bv;
    // G2[h][w] = exp(-(x[h][w-1]-x[h][w])^2)+b    (w>=1)
    o2.x = wok0 ? GATE(cm1, cv.x) : bv;
    o2.y =        GATE(cv.x, cv.y);
    o2.z =        GATE(cv.y, cv.z);
    o2.w =        GATE(cv.z, cv.w);
    // G3[h][w] = exp(-(x[h+1][w-1]-x[h][w])^2)+b  (h<=510, w>=1)
    o3.x = (hn && wok0) ? GATE(nm1,  cv.x) : bv;
    o3.y =  hn          ? GATE(nv.x, cv.y) : bv;
    o3.z =  hn          ? GATE(nv.y, cv.z) : bv;
    o3.w =  hn          ? GATE(nv.z, cv.w) : bv;
#undef GATE

    // Write-once outputs (384 MiB total > L2): non-temporal b128 stores.
    const long long N   = (long long)NPLANES * W * W;       // 33554432
    const long long off = (long long)(r0 + lr) * W + w0;
    __builtin_nontemporal_store(o1, (f32x4*)(out + off));
    __builtin_nontemporal_store(o2, (f32x4*)(out + N + off));
    __builtin_nontemporal_store(o3, (f32x4*)(out + 2 * N + off));
}

extern "C" void kernel_launch(void* const* d_in, const int* in_sizes, int n_in,
                              void* d_out, int out_size, void* d_ws, size_t ws_size,
                              hipStream_t stream) {
    const float* x    = (const float*)d_in[0];   // (4,32,512,512) fp32
    const float* bias = (const float*)d_in[1];   // (1,32,1,1) -> 32 fp32
    float*       out  = (float*)d_out;           // 3 * 33554432 fp32

    const int totalRows = NPLANES * W;           // 65536
    dim3 grid(totalRows / ROWS_OUT);             // 32768 workgroups
    dim3 block(256);                             // 8 waves (wave32)
    embeded_gate_tdm_kernel<<<grid, block, 0, stream>>>(x, bias, out);
}